// VecInt_58961311040341
// MI455X (gfx1250) — compile-verified
//
#include <hip/hip_runtime.h>
#include <cstdint>

// ---------------------------------------------------------------------------
// VecInt (scaling-and-squaring integration of a 3D velocity field)
//   vec: (1, 3, 160, 192, 160) fp32, nsteps = 7
//   step: out = v + trilinear_sample(v, v + identity_grid)   (zero padding)
// L2-resident ping-pong (2 x 59 MB < 192 MB L2) -> optimize for L2 bandwidth
// and per-wave latency: 3D launch (no index-division on the critical path),
// coalesced streams, clamped gather addresses, border validity folded into
// the interpolation weights (w=0 <=> value=0), FMA accumulation.
// CDNA5 path: async global->LDS staging of the streaming center loads
// (global_load_async_to_lds_b32 + s_wait_asynccnt). WMMA is inapplicable:
// trilinear sampling is a per-voxel diagonal operator, no matmul structure.
// ---------------------------------------------------------------------------

#ifndef __has_builtin
#define __has_builtin(x) 0
#endif

#if __has_builtin(__builtin_amdgcn_global_load_async_to_lds_b32)
#define HAVE_ASYNC_LDS 1
#else
#define HAVE_ASYNC_LDS 0
#endif

#define AS1 __attribute__((address_space(1)))
#define AS3 __attribute__((address_space(3)))

static constexpr int Dd   = 160;
static constexpr int Hh   = 192;
static constexpr int Ww   = 160;
static constexpr int NVOX = Dd * Hh * Ww;       // 4,915,200
static constexpr int NSTEPS = 7;                // matches setup_inputs()
static constexpr int BLOCK = Ww;                // one x-row = 160 = 5 wave32s

__global__ __launch_bounds__(BLOCK) void vecint_step_kernel(
    const float* __restrict__ src, float* __restrict__ dst, float prescale)
{
    // 3D launch: no integer division anywhere.
    const int x = threadIdx.x;          // 0..159
    const int y = blockIdx.x;           // 0..191
    const int z = blockIdx.y;           // 0..159
    const int i = (z * Hh + y) * Ww + x;

    // ---- streaming center load of all 3 channels ------------------------
    float c0, c1, c2;
#if HAVE_ASYNC_LDS
    __shared__ float cbuf[3 * BLOCK];
    {
        AS1 int* g0 = (AS1 int*)(uintptr_t)(src + 0 * NVOX + i);
        AS1 int* g1 = (AS1 int*)(uintptr_t)(src + 1 * NVOX + i);
        AS1 int* g2 = (AS1 int*)(uintptr_t)(src + 2 * NVOX + i);
        AS3 int* l0 = (AS3 int*)(uintptr_t)(&cbuf[0 * BLOCK + x]);
        AS3 int* l1 = (AS3 int*)(uintptr_t)(&cbuf[1 * BLOCK + x]);
        AS3 int* l2 = (AS3 int*)(uintptr_t)(&cbuf[2 * BLOCK + x]);
        __builtin_amdgcn_global_load_async_to_lds_b32(g0, l0, 0, 0);
        __builtin_amdgcn_global_load_async_to_lds_b32(g1, l1, 0, 0);
        __builtin_amdgcn_global_load_async_to_lds_b32(g2, l2, 0, 0);
#if __has_builtin(__builtin_amdgcn_s_wait_asynccnt)
        __builtin_amdgcn_s_wait_asynccnt(0);
#else
        asm volatile("s_wait_asynccnt 0" ::: "memory");
#endif
        // each lane reads back only its own slot: no cross-lane barrier needed
        c0 = cbuf[0 * BLOCK + x];
        c1 = cbuf[1 * BLOCK + x];
        c2 = cbuf[2 * BLOCK + x];
    }
#else
    c0 = src[0 * NVOX + i];
    c1 = src[1 * NVOX + i];
    c2 = src[2 * NVOX + i];
#endif

    // step-1 fuses the 2^-NSTEPS pre-scale (prescale), identity otherwise
    c0 *= prescale;  c1 *= prescale;  c2 *= prescale;

    // sample position in voxel units:  p = voxel + v * 0.5*(dim-1)
    const float pz = fmaf(c0, 0.5f * (Dd - 1), (float)z);
    const float py = fmaf(c1, 0.5f * (Hh - 1), (float)y);
    const float px = fmaf(c2, 0.5f * (Ww - 1), (float)x);

    const float zf = floorf(pz), yf = floorf(py), xf = floorf(px);
    const float fz = pz - zf,    fy = py - yf,    fx = px - xf;
    const int z0 = (int)zf, y0 = (int)yf, x0 = (int)xf;
    const int z1 = z0 + 1,  y1 = y0 + 1,  x1 = x0 + 1;

    // zero-padding: fold per-axis validity into the weights (w*0 == 0*v)
    float wz0 = (z0 >= 0 && z0 < Dd) ? (1.0f - fz) : 0.0f;
    float wz1 = (z1 >= 0 && z1 < Dd) ? fz          : 0.0f;
    const float wy0 = (y0 >= 0 && y0 < Hh) ? (1.0f - fy) : 0.0f;
    const float wy1 = (y1 >= 0 && y1 < Hh) ? fy          : 0.0f;
    const float wx0 = (x0 >= 0 && x0 < Ww) ? (1.0f - fx) : 0.0f;
    const float wx1 = (x1 >= 0 && x1 < Ww) ? fx          : 0.0f;
    // fold gather pre-scale into one weight axis
    wz0 *= prescale;
    wz1 *= prescale;

    // clamped corner indices (loads always in-bounds; bad corners have w==0)
    const int zc0 = min(max(z0, 0), Dd - 1), zc1 = min(max(z1, 0), Dd - 1);
    const int yc0 = min(max(y0, 0), Hh - 1), yc1 = min(max(y1, 0), Hh - 1);
    const int xc0 = min(max(x0, 0), Ww - 1), xc1 = min(max(x1, 0), Ww - 1);

    const int o00 = (zc0 * Hh + yc0) * Ww;
    const int o01 = (zc0 * Hh + yc1) * Ww;
    const int o10 = (zc1 * Hh + yc0) * Ww;
    const int o11 = (zc1 * Hh + yc1) * Ww;

    const float* __restrict__ s1p = src + NVOX;
    const float* __restrict__ s2p = src + 2 * NVOX;

    float a0 = c0, a1 = c1, a2 = c2;    // out = center + sum(w * gathered)
#define CORNER(OFF, WGT)                                        \
    do {                                                        \
        const int   _o = (OFF);                                 \
        const float _w = (WGT);                                 \
        a0 = fmaf(_w, src[_o], a0);                             \
        a1 = fmaf(_w, s1p[_o], a1);                             \
        a2 = fmaf(_w, s2p[_o], a2);                             \
    } while (0)

    CORNER(o00 + xc0, wz0 * wy0 * wx0);
    CORNER(o00 + xc1, wz0 * wy0 * wx1);
    CORNER(o01 + xc0, wz0 * wy1 * wx0);
    CORNER(o01 + xc1, wz0 * wy1 * wx1);
    CORNER(o10 + xc0, wz1 * wy0 * wx0);
    CORNER(o10 + xc1, wz1 * wy0 * wx1);
    CORNER(o11 + xc0, wz1 * wy1 * wx0);
    CORNER(o11 + xc1, wz1 * wy1 * wx1);
#undef CORNER

    dst[0 * NVOX + i] = a0;
    dst[1 * NVOX + i] = a1;
    dst[2 * NVOX + i] = a2;
}

extern "C" void kernel_launch(void* const* d_in, const int* in_sizes, int n_in,
                              void* d_out, int out_size, void* d_ws, size_t ws_size,
                              hipStream_t stream)
{
    (void)in_sizes; (void)n_in; (void)out_size; (void)ws_size;

    const float* vec = (const float*)d_in[0];
    float* out = (float*)d_out;
    float* ws  = (float*)d_ws;          // needs >= 3*NVOX*4 = ~59 MB

    const float scale = 1.0f / (float)(1 << NSTEPS);   // 2^-7
    const dim3 grid(Hh, Dd), block(BLOCK);             // (y, z), x in block

    // 7 steps, ping-pong ending in d_out; step 1 fuses the 2^-7 pre-scale.
    vecint_step_kernel<<<grid, block, 0, stream>>>(vec, out, scale);   // 1: in -> out
    vecint_step_kernel<<<grid, block, 0, stream>>>(out, ws, 1.0f);     // 2: out -> ws
    vecint_step_kernel<<<grid, block, 0, stream>>>(ws,  out, 1.0f);    // 3: ws  -> out
    vecint_step_kernel<<<grid, block, 0, stream>>>(out, ws, 1.0f);     // 4: out -> ws
    vecint_step_kernel<<<grid, block, 0, stream>>>(ws,  out, 1.0f);    // 5: ws  -> out
    vecint_step_kernel<<<grid, block, 0, stream>>>(out, ws, 1.0f);     // 6: out -> ws
    vecint_step_kernel<<<grid, block, 0, stream>>>(ws,  out, 1.0f);    // 7: ws  -> out
}